// MSDeformAttn_2319282340121
// MI455X (gfx1250) — compile-verified
//
#include <hip/hip_runtime.h>
#include <hip/hip_bf16.h>

// ---- Problem constants (from reference) ----
#define D_MODEL   256
#define N_HEADS   8
#define HEAD_DIM  32
#define N_LEVELS  4
#define N_POINTS  4
#define LEN_IN_C  22223   // 100*167 + 50*84 + 25*42 + 13*21
#define LEN_Q_C   22223
#define NBATCH    2

typedef __attribute__((ext_vector_type(2))) float v2f;
typedef __attribute__((ext_vector_type(8))) float v8f;

// ---------------------------------------------------------------------------
// C[R x NCOLS] = A[R x 256] * B[256 x NCOLS] + bias[NCOLS]
// fp32 WMMA 16x16x4. One wave computes a 64-row x 16-col strip (4 M-tiles),
// reusing the B fragment across 4 WMMAs per k-step. 8 waves per block.
//
// A-fragment (32-bit A 16x4): lanes 0-15 row M=lane, regs hold K=0,1;
// lanes 16-31 row M=lane-16, regs hold K=2,3.  B mirrored (4x16).
// C/D (32-bit 16x16): VGPR r -> M=r (lanes 0-15), M=8+r (lanes 16-31).
// ---------------------------------------------------------------------------
template <int NCOLS>
__global__ __launch_bounds__(256) void gemm_bias_f32(
    const float* __restrict__ A, const float* __restrict__ B,
    const float* __restrict__ bias, float* __restrict__ C, int R) {
  constexpr int K  = D_MODEL;
  constexpr int TN = NCOLS / 16;       // col tiles
  const int wave = threadIdx.x >> 5;
  const int lane = threadIdx.x & 31;
  const long g    = (long)blockIdx.x * 8 + wave;  // wave-strip id
  const int  gm   = (int)(g / TN);                // 64-row group
  const int  gn   = (int)(g % TN);                // 16-col tile
  const int  row0 = gm * 64;
  if (row0 >= R) return;               // wave-uniform

  const int lh = lane & 15;
  const int kh = (lane >> 4) << 1;     // 0 for lanes 0-15, 2 for lanes 16-31

  const float* Ap[4];
#pragma unroll
  for (int t = 0; t < 4; ++t) {
    int r = row0 + t * 16 + lh;
    if (r >= R) r = R - 1;             // clamp; rows >= R never stored
    Ap[t] = A + (long)r * K + kh;      // 8B aligned (kh even)
  }
  const float* Bp = B + (long)kh * NCOLS + gn * 16 + lh;

  v8f acc[4] = {};
#pragma unroll 4
  for (int k = 0; k < K; k += 4) {
    v2f b;
    b.x = Bp[(long)k * NCOLS];         // K = k+kh
    b.y = Bp[(long)(k + 1) * NCOLS];   // K = k+kh+1
    v2f a0 = *(const v2f*)(Ap[0] + k);
    v2f a1 = *(const v2f*)(Ap[1] + k);
    v2f a2 = *(const v2f*)(Ap[2] + k);
    v2f a3 = *(const v2f*)(Ap[3] + k);
    acc[0] = __builtin_amdgcn_wmma_f32_16x16x4_f32(false, a0, false, b, (short)0, acc[0], false, false);
    acc[1] = __builtin_amdgcn_wmma_f32_16x16x4_f32(false, a1, false, b, (short)0, acc[1], false, false);
    acc[2] = __builtin_amdgcn_wmma_f32_16x16x4_f32(false, a2, false, b, (short)0, acc[2], false, false);
    acc[3] = __builtin_amdgcn_wmma_f32_16x16x4_f32(false, a3, false, b, (short)0, acc[3], false, false);
  }

  const int   ccol = gn * 16 + lh;
  const float bv   = bias[ccol];
  const int   rbase = row0 + ((lane >> 4) << 3);  // +8 for upper half-lanes

  if (row0 + 64 <= R) {                // wave-uniform fast path: full strip
#pragma unroll
    for (int t = 0; t < 4; ++t)
#pragma unroll
      for (int r = 0; r < 8; ++r)
        C[(long)(rbase + t * 16 + r) * NCOLS + ccol] = acc[t][r] + bv;
  } else {                             // tail strip: predicated stores
#pragma unroll
    for (int t = 0; t < 4; ++t)
#pragma unroll
      for (int r = 0; r < 8; ++r) {
        const int crow = rbase + t * 16 + r;
        if (crow < R) C[(long)crow * NCOLS + ccol] = acc[t][r] + bv;
      }
  }
}

// ---------------------------------------------------------------------------
// Deformable sampling: one thread per (n, q, head, 4-channel group).
// Softmax(16) + bilinear gather of float4 channel groups, accumulate L*P=16.
// value tensor (45 MB) fits in the 192 MB L2, so gathers are L2 hits.
// ---------------------------------------------------------------------------
__global__ __launch_bounds__(256) void msda_sample(
    const float* __restrict__ value,     // (N, LEN_IN, 256) = (n, pix, m*32+c)
    const float* __restrict__ off_raw,   // (N*LQ, 256) = (m, l, p, 2)
    const float* __restrict__ attn_raw,  // (N*LQ, 128) = (m, l, p)
    const float* __restrict__ ref,       // (N*LQ, 4, 2)
    const int*   __restrict__ shapes,    // (4,2) H,W
    const int*   __restrict__ starts,    // (4,)
    float* __restrict__ out_tmp) {       // (N*LQ, 256)
  const long gid   = (long)blockIdx.x * blockDim.x + threadIdx.x;
  const long total = (long)NBATCH * LEN_Q_C * N_HEADS * (HEAD_DIM / 4);
  if (gid >= total) return;

  const int  cg    = (int)(gid & 7);          // channel group 0..7
  const int  m     = (int)((gid >> 3) & 7);   // head
  const long nq    = gid >> 6;                // n*LQ + q
  const int  n     = (int)(nq / LEN_Q_C);
  const int  cbase = m * HEAD_DIM + cg * 4;

  // softmax over the 16 (level,point) logits of this head
  const float* al = attn_raw + nq * (N_HEADS * N_LEVELS * N_POINTS)
                             + m * (N_LEVELS * N_POINTS);
  float mx = al[0];
#pragma unroll
  for (int i = 1; i < 16; ++i) mx = fmaxf(mx, al[i]);
  float w[16];
  float sum = 0.f;
#pragma unroll
  for (int i = 0; i < 16; ++i) { w[i] = __expf(al[i] - mx); sum += w[i]; }
  const float inv = 1.f / sum;

  const float* ofs = off_raw + nq * (N_HEADS * N_LEVELS * N_POINTS * 2)
                             + m * (N_LEVELS * N_POINTS * 2);
  const float* rp  = ref + nq * (N_LEVELS * 2);

  float a0 = 0.f, a1 = 0.f, a2 = 0.f, a3 = 0.f;

  for (int l = 0; l < N_LEVELS; ++l) {
    const int   H  = shapes[l * 2 + 0];
    const int   W  = shapes[l * 2 + 1];
    const int   s  = starts[l];
    const float fw = (float)W, fh = (float)H;
    const float rx = rp[l * 2 + 0], ry = rp[l * 2 + 1];
    const float* vl = value + ((long)n * LEN_IN_C + s) * D_MODEL + cbase;

#pragma unroll
    for (int p = 0; p < N_POINTS; ++p) {
      const float ox = ofs[(l * N_POINTS + p) * 2 + 0];
      const float oy = ofs[(l * N_POINTS + p) * 2 + 1];
      const float aw = w[l * N_POINTS + p] * inv;
      // loc = ref + off/[W,H];  x = loc_x*W - 0.5 = rx*W + ox - 0.5
      const float x  = rx * fw + ox - 0.5f;
      const float y  = ry * fh + oy - 0.5f;
      const float x0 = floorf(x), y0 = floorf(y);
      const float lx = x - x0,    ly = y - y0;
      const int   ix = (int)x0,   iy = (int)y0;

      float s0 = 0.f, s1 = 0.f, s2 = 0.f, s3 = 0.f;
      const float cw[4] = { (1.f - lx) * (1.f - ly), lx * (1.f - ly),
                            (1.f - lx) * ly,         lx * ly };
      const int cx[4] = { ix, ix + 1, ix,     ix + 1 };
      const int cy[4] = { iy, iy,     iy + 1, iy + 1 };
#pragma unroll
      for (int c = 0; c < 4; ++c) {
        if (cx[c] >= 0 && cx[c] < W && cy[c] >= 0 && cy[c] < H) {
          const float4 g =
              *(const float4*)(vl + (long)(cy[c] * W + cx[c]) * D_MODEL);
          s0 += cw[c] * g.x; s1 += cw[c] * g.y;
          s2 += cw[c] * g.z; s3 += cw[c] * g.w;
        }
      }
      a0 += aw * s0; a1 += aw * s1; a2 += aw * s2; a3 += aw * s3;
    }
  }

  float4 o; o.x = a0; o.y = a1; o.z = a2; o.w = a3;
  *(float4*)(out_tmp + nq * D_MODEL + cbase) = o;
}

// ---------------------------------------------------------------------------
extern "C" void kernel_launch(void* const* d_in, const int* in_sizes, int n_in,
                              void* d_out, int out_size, void* d_ws, size_t ws_size,
                              hipStream_t stream) {
  const float* query   = (const float*)d_in[0];
  const float* refp    = (const float*)d_in[1];
  const float* inpf    = (const float*)d_in[2];
  const int*   shapes  = (const int*)d_in[3];
  const int*   starts  = (const int*)d_in[4];
  const float* w_value = (const float*)d_in[5];
  const float* b_value = (const float*)d_in[6];
  const float* w_off   = (const float*)d_in[7];
  const float* b_off   = (const float*)d_in[8];
  const float* w_attn  = (const float*)d_in[9];
  const float* b_attn  = (const float*)d_in[10];
  const float* w_out   = (const float*)d_in[11];
  const float* b_out   = (const float*)d_in[12];
  float* out = (float*)d_out;

  const int  R = NBATCH * LEN_Q_C;        // 44446 rows for every GEMM
  const long V = (long)R * D_MODEL;       // 11.38M elements (45.5 MB)

  // workspace layout: value | off_raw | tmp | attn_raw  (~160 MB total)
  float* value    = (float*)d_ws;
  float* off_raw  = value + V;
  float* tmp      = off_raw + V;
  float* attn_raw = tmp + V;

  const int gm  = (R + 63) / 64;          // 695 64-row groups
  dim3 blk(256);
  const int g256 = (gm * 16 + 7) / 8;     // blocks, NCOLS=256 (8 waves/blk)
  const int g128 = (gm * 8 + 7) / 8;      // blocks, NCOLS=128

  // 1) value projection
  gemm_bias_f32<256><<<g256, blk, 0, stream>>>(inpf, w_value, b_value, value, R);
  // 2) sampling offsets (raw)
  gemm_bias_f32<256><<<g256, blk, 0, stream>>>(query, w_off, b_off, off_raw, R);
  // 3) attention logits (raw)
  gemm_bias_f32<128><<<g128, blk, 0, stream>>>(query, w_attn, b_attn, attn_raw, R);
  // 4) softmax + bilinear sampling + head merge
  const long total = (long)R * N_HEADS * (HEAD_DIM / 4);
  msda_sample<<<(int)((total + 255) / 256), blk, 0, stream>>>(
      value, off_raw, attn_raw, refp, shapes, starts, tmp);
  // 5) output projection
  gemm_bias_f32<256><<<g256, blk, 0, stream>>>(tmp, w_out, b_out, out, R);
}